// Encoder_67912022884450
// MI455X (gfx1250) — compile-verified
//
#include <hip/hip_runtime.h>

typedef __attribute__((ext_vector_type(2))) float v2f;
typedef __attribute__((ext_vector_type(8))) float v8f;

#define N_NODES 50000
#define IN_CH   128
#define HID_CH  128
#define OUT_CH  64

// ---------------------------------------------------------------- utilities

__global__ void zero_f4_kernel(float4* __restrict__ p, long n4) {
    long t = (long)blockIdx.x * blockDim.x + threadIdx.x;
    if (t < n4) p[t] = make_float4(0.f, 0.f, 0.f, 0.f);
}

__global__ void zero_f_kernel(float* __restrict__ p, long n) {
    long t = (long)blockIdx.x * blockDim.x + threadIdx.x;
    if (t < n) p[t] = 0.f;
}

// ------------------------------------------------------------ degree / norm

__global__ void deg_count_kernel(const int* __restrict__ dst,
                                 float* __restrict__ deg, int E) {
    long t = (long)blockIdx.x * blockDim.x + threadIdx.x;
    if (t < E) unsafeAtomicAdd(&deg[dst[t]], 1.0f);
}

__global__ void deg_finish_kernel(const float* __restrict__ deg,
                                  float* __restrict__ dis, int N) {
    long t = (long)blockIdx.x * blockDim.x + threadIdx.x;
    if (t < N) dis[t] = rsqrtf(deg[t] + 1.0f);   // +1 = self loop; always > 0
}

// ------------------------------------------------------- f32 WMMA dense GEMM
// C[M,N] = A[M,K] @ B[K,N], all f32, M%16==0, N%16==0, K%4==0.
// One wave computes one 16x16 tile via V_WMMA_F32_16X16X4_F32.
//
// A frag (16x4, ISA layout): lanes 0-15 hold rows M=lane with K={0,1} in the
// two VGPRs; lanes 16-31 hold the same rows with K={2,3}.
// B frag (4x16): symmetric — lane column n=lane&15, K pair selected by lane[4].
// C/D (16x16 f32, 8 VGPRs): VGPR v, lanes 0-15 -> row v, lanes 16-31 -> row v+8.

__global__ void gemm_wmma_f32_kernel(const float* __restrict__ A,
                                     const float* __restrict__ B,
                                     float* __restrict__ C,
                                     int M, int K, int N) {
    const int lane  = threadIdx.x & 31;
    const int wib   = threadIdx.x >> 5;                 // wave in block
    const int wpb   = blockDim.x >> 5;                  // waves per block
    const int wave  = blockIdx.x * wpb + wib;

    const int tilesN = N >> 4;
    const int totalTiles = (M >> 4) * tilesN;
    if (wave >= totalTiles) return;                     // wave-uniform exit

    const int tm = wave / tilesN;
    const int tn = wave - tm * tilesN;
    const int m0 = tm << 4;
    const int n0 = tn << 4;

    const int mi = lane & 15;                           // A row / B col in tile
    const int kk = (lane >> 4) << 1;                    // 0 or 2: K sub-pair

    v8f acc = {0.f, 0.f, 0.f, 0.f, 0.f, 0.f, 0.f, 0.f};

    const float* __restrict__ arow = A + (size_t)(m0 + mi) * K;

    for (int k0 = 0; k0 < K; k0 += 4) {
        v2f a;
        a.x = arow[k0 + kk];
        a.y = arow[k0 + kk + 1];
        v2f b;
        b.x = B[(size_t)(k0 + kk)     * N + n0 + mi];
        b.y = B[(size_t)(k0 + kk + 1) * N + n0 + mi];
        // 8 args: (neg_a, A, neg_b, B, c_mod, C, reuse_a, reuse_b)
        acc = __builtin_amdgcn_wmma_f32_16x16x4_f32(
            false, a, false, b, (short)0, acc, false, false);
    }

    const int rbase = m0 + ((lane >> 4) << 3);          // +0 or +8
    const int col   = n0 + mi;
#pragma unroll
    for (int v = 0; v < 8; ++v)
        C[(size_t)(rbase + v) * N + col] = acc[v];
}

// --------------------------------------------------------- edge scatter-add
// One thread per (edge, 4-channel group): float4 gather from h[src], scale by
// dis[src]*dis[dst], 4 native f32 atomics into agg[dst]. h fits in L2 so the
// random gathers are L2 hits; atomics are RMW in L2.

__global__ void scatter_edges_kernel(const float* __restrict__ h,
                                     const int* __restrict__ src,
                                     const int* __restrict__ dst,
                                     const float* __restrict__ dis,
                                     float* __restrict__ agg,
                                     int E, int C4) {
    long t = (long)blockIdx.x * blockDim.x + threadIdx.x;
    long total = (long)E * C4;
    if (t >= total) return;
    int e = (int)(t / C4);
    int g = (int)(t - (long)e * C4);
    int s = src[e], d = dst[e];
    float nrm = dis[s] * dis[d];
    const int C = C4 << 2;
    float4 v = ((const float4*)(h + (size_t)s * C))[g];
    float* ad = agg + (size_t)d * C + (g << 2);
    unsafeAtomicAdd(ad + 0, v.x * nrm);
    unsafeAtomicAdd(ad + 1, v.y * nrm);
    unsafeAtomicAdd(ad + 2, v.z * nrm);
    unsafeAtomicAdd(ad + 3, v.w * nrm);
}

// -------------------------------------------- self-loop + bias + ReLU epilogue
// out[i,c] = relu( agg[i,c] + dis[i]^2 * hlin[i,c] + bias[c] )
// (self-loop term dis^2*h folded here instead of atomics)

__global__ void bias_relu_kernel(const float* __restrict__ agg,
                                 const float* __restrict__ hlin,
                                 const float* __restrict__ dis,
                                 const float* __restrict__ bias,
                                 float* __restrict__ outp,
                                 int N, int C) {
    long t = (long)blockIdx.x * blockDim.x + threadIdx.x;
    long total = (long)N * C;
    if (t >= total) return;
    int i = (int)(t / C);
    int c = (int)(t - (long)i * C);
    float ds = dis[i];
    float v = agg[t] + ds * ds * hlin[t] + bias[c];
    outp[t] = v > 0.f ? v : 0.f;
}

// ------------------------------------------------------------------- launch

static inline long cdiv(long a, long b) { return (a + b - 1) / b; }

extern "C" void kernel_launch(void* const* d_in, const int* in_sizes, int n_in,
                              void* d_out, int out_size, void* d_ws, size_t ws_size,
                              hipStream_t stream) {
    const float* x  = (const float*)d_in[0];            // [N,128]
    const int*   ei = (const int*)d_in[1];              // [2,E]
    const float* W1 = (const float*)d_in[2];            // [128,128]
    const float* b1 = (const float*)d_in[3];            // [128]
    const float* W2 = (const float*)d_in[4];            // [128,64]
    const float* b2 = (const float*)d_in[5];            // [64]
    float* out = (float*)d_out;                         // [N,64]

    const int N = N_NODES;
    const int E = in_sizes[1] / 2;
    const int* src = ei;
    const int* dst = ei + E;

    // ---- workspace carve (256B aligned) ----
    char* ws = (char*)d_ws;
    size_t off = 0;
    auto alloc = [&](size_t bytes) {
        char* p = ws + off;
        off += (bytes + 255) & ~(size_t)255;
        return p;
    };
    float* deg  = (float*)alloc((size_t)N * sizeof(float));
    float* dis  = (float*)alloc((size_t)N * sizeof(float));
    float* h1   = (float*)alloc((size_t)N * HID_CH * sizeof(float)); // X@W1, then relu'd h
    float* agg1 = (float*)alloc((size_t)N * HID_CH * sizeof(float)); // layer1 agg, then reused as h2
    float* h2   = agg1;                                              // [N,64] fits in [N,128]
    (void)ws_size; (void)n_in; (void)out_size;

    const int TB = 256;

    // ---- degree / normalization ----
    zero_f_kernel<<<cdiv(N, TB), TB, 0, stream>>>(deg, N);
    deg_count_kernel<<<cdiv(E, TB), TB, 0, stream>>>(dst, deg, E);
    deg_finish_kernel<<<cdiv(N, TB), TB, 0, stream>>>(deg, dis, N);

    // ---- layer 1: h1 = X @ W1 ----
    {
        long tiles = (long)(N / 16) * (HID_CH / 16);    // 3125 * 8
        long blocks = cdiv(tiles, TB / 32);
        gemm_wmma_f32_kernel<<<blocks, TB, 0, stream>>>(x, W1, h1, N, IN_CH, HID_CH);
    }
    zero_f4_kernel<<<cdiv((long)N * HID_CH / 4, TB), TB, 0, stream>>>(
        (float4*)agg1, (long)N * HID_CH / 4);
    scatter_edges_kernel<<<cdiv((long)E * (HID_CH / 4), TB), TB, 0, stream>>>(
        h1, src, dst, dis, agg1, E, HID_CH / 4);
    bias_relu_kernel<<<cdiv((long)N * HID_CH, TB), TB, 0, stream>>>(
        agg1, h1, dis, b1, h1, N, HID_CH);              // h (relu) written into h1

    // ---- layer 2: h2 = h @ W2 ----
    {
        long tiles = (long)(N / 16) * (OUT_CH / 16);    // 3125 * 4
        long blocks = cdiv(tiles, TB / 32);
        gemm_wmma_f32_kernel<<<blocks, TB, 0, stream>>>(h1, W2, h2, N, HID_CH, OUT_CH);
    }
    zero_f4_kernel<<<cdiv((long)N * OUT_CH / 4, TB), TB, 0, stream>>>(
        (float4*)out, (long)N * OUT_CH / 4);
    scatter_edges_kernel<<<cdiv((long)E * (OUT_CH / 4), TB), TB, 0, stream>>>(
        h2, src, dst, dis, out, E, OUT_CH / 4);
    bias_relu_kernel<<<cdiv((long)N * OUT_CH, TB), TB, 0, stream>>>(
        out, h2, dis, b2, out, N, OUT_CH);
}